// Pool_conv_sum_nonlin_pool_41317585388092
// MI455X (gfx1250) — compile-verified
//
#include <hip/hip_runtime.h>
#include <cfloat>
#include <cstddef>

#define B_ 4
#define C_ 16
#define O_ 32
#define P_ 256
#define G_ 2048

typedef __attribute__((ext_vector_type(2))) float v2f;
typedef __attribute__((ext_vector_type(8))) float v8f;

// Order-preserving float -> u32 key (monotone: a<b  <=>  key(a)<key(b)).
// key(x) > 0 for all finite/inf x, so 0 is a safe "minus infinity" init.
__device__ __forceinline__ unsigned f2key(float x) {
  unsigned b = __float_as_uint(x);
  return (b & 0x80000000u) ? ~b : (b | 0x80000000u);
}
__device__ __forceinline__ float key2f(unsigned k) {
  unsigned b = (k & 0x80000000u) ? (k & 0x7FFFFFFFu) : ~k;
  return __uint_as_float(b);
}

// leaky_relu(x) with slope 0.01 < 1 is exactly max(x, 0.01*x):
// x>=0 -> x >= 0.01x ; x<0 -> 0.01x > x.  One v_max_num instead of cmp+cndmask.
__device__ __forceinline__ float leaky(float x) {
  return fmaxf(x, 0.01f * x);
}

// ---------------- kernel 0: init out_p keys ----------------
__global__ void init_keys_k(unsigned* __restrict__ keys) {
  keys[blockIdx.x * 256 + threadIdx.x] = 0u;   // below any real value
}

// ---------------- kernel 1: pool_g = max over g ----------------
// one wave per (b,c,p) row of G=2048 floats; 8 waves per block
__global__ __launch_bounds__(256) void pool_g_k(const float* __restrict__ x,
                                                float* __restrict__ pg) {
  int row  = blockIdx.x * 8 + (threadIdx.x >> 5);   // < B*C*P = 16384
  int lane = threadIdx.x & 31;
  const float* src = x + (size_t)row * G_;
  float m = -FLT_MAX;
  for (int g = lane; g < G_; g += 32) m = fmaxf(m, src[g]);
#pragma unroll
  for (int mk = 16; mk >= 1; mk >>= 1) m = fmaxf(m, __shfl_xor(m, mk, 32));
  if (lane == 0) pg[row] = m;
}

// ---------------- kernel 2: pool_p = max over p ----------------
// one thread per (b,c,g); consecutive threads -> consecutive g (coalesced)
__global__ void pool_p_k(const float* __restrict__ x, float* __restrict__ pp) {
  int tid = blockIdx.x * 256 + threadIdx.x;         // < B*C*G = 131072
  int bc = tid / G_;
  int g  = tid % G_;
  const float* src = x + (size_t)bc * P_ * G_ + g;
  float m = -FLT_MAX;
  for (int p = 0; p < P_; ++p) m = fmaxf(m, src[(size_t)p * G_]);
  pp[tid] = m;
}

// ---------------- kernel 3: y_p (B,O,P) ----------------
__global__ void yp_k(const float* __restrict__ pool_g, const float* __restrict__ x_p,
                     const float* __restrict__ W_pg_p, const float* __restrict__ b_pg_p,
                     const float* __restrict__ W_p,    const float* __restrict__ b_p,
                     float* __restrict__ y_p) {
  int tid = blockIdx.x * 256 + threadIdx.x;         // < B*O*P = 32768
  int p = tid % P_;
  int o = (tid / P_) % O_;
  int b = tid / (P_ * O_);
  float s = b_pg_p[o] + b_p[o];
#pragma unroll
  for (int c = 0; c < C_; ++c) {
    int bc = b * C_ + c;
    s += W_pg_p[o * C_ + c] * pool_g[bc * P_ + p]
       + W_p[o * C_ + c]    * x_p[bc * P_ + p];
  }
  y_p[tid] = s;
}

// ---------------- kernel 4: y_g (B,O,G) ----------------
__global__ void yg_k(const float* __restrict__ pool_p, const float* __restrict__ x_g,
                     const float* __restrict__ W_pg_g, const float* __restrict__ b_pg_g,
                     const float* __restrict__ W_g,    const float* __restrict__ b_g,
                     float* __restrict__ y_g) {
  int tid = blockIdx.x * 256 + threadIdx.x;         // < B*O*G = 262144
  int g = tid % G_;
  int o = (tid / G_) % O_;
  int b = tid / (G_ * O_);
  float s = b_pg_g[o] + b_g[o];
#pragma unroll
  for (int c = 0; c < C_; ++c) {
    int bc = b * C_ + c;
    s += W_pg_g[o * C_ + c] * pool_p[bc * G_ + g]
       + W_g[o * C_ + c]    * x_g[bc * G_ + g];
  }
  y_g[tid] = s;
}

// ---------------- kernel 5: main fused WMMA kernel ----------------
// grid = B * (G/16); each block owns a 16-wide g tile; 8 waves split p.
// Per wave-iteration: D(32x16) = W_pg_pg(32x16) x X(16x16) via
// 2 M-tiles x 4 chained v_wmma_f32_16x16x4_f32, then fused epilogue.
__global__ __launch_bounds__(256) void main_wmma_k(
    const float* __restrict__ x_pg, const float* __restrict__ Wpg,
    const float* __restrict__ bpg,  const float* __restrict__ y_p,
    const float* __restrict__ y_g,  float* __restrict__ out_pg,
    unsigned* __restrict__ outp_keys, float* __restrict__ out_g) {
  const int NT = G_ / 16;                       // 128 g-tiles
  int b    = blockIdx.x / NT;
  int g0   = (blockIdx.x % NT) * 16;
  int lane = threadIdx.x & 31;
  int wave = threadIdx.x >> 5;                  // 0..7
  int hf   = lane >> 4;                         // half-wave: 0 or 1
  int n    = lane & 15;                         // N position / M row id

  // A fragments: W_pg_pg row-major (O=32, C=16).
  // 16x4 f32 A layout: lanes0-15 hold M=n with K={0,1}, lanes16-31 K={2,3}.
  v2f afrag[2][4];
#pragma unroll
  for (int t = 0; t < 2; ++t)
#pragma unroll
    for (int kc = 0; kc < 4; ++kc) {
      const float* wp = Wpg + (n + 16 * t) * C_ + 4 * kc + 2 * hf;
      afrag[t][kc].x = wp[0];
      afrag[t][kc].y = wp[1];
    }

  // Hoist bias + y_g (both constant over p).  C/D layout: VGPR r ->
  // M = r + 8*hf (+16 per tile), N = n.
  float cadd[2][8];
#pragma unroll
  for (int t = 0; t < 2; ++t)
#pragma unroll
    for (int r = 0; r < 8; ++r) {
      int o = 16 * t + 8 * hf + r;
      cadd[t][r] = bpg[o] + y_g[(size_t)(b * O_ + o) * G_ + g0 + n];
    }

  float ogmax[2][8];
#pragma unroll
  for (int t = 0; t < 2; ++t)
#pragma unroll
    for (int r = 0; r < 8; ++r) ogmax[t][r] = -FLT_MAX;

  // After the butterfly tree, lane j (in its 16-group) holds the full
  // 16-lane reduction of register index bitrev4(j); that register maps to
  // output channel o_red.  One non-divergent atomic per lane.
  int idxr  = ((n & 1) << 3) | ((n & 2) << 1) | ((n & 4) >> 1) | ((n & 8) >> 3);
  int o_red = 16 * (idxr >> 3) + 8 * hf + (idxr & 7);
  unsigned* opk_base = outp_keys + (size_t)(b * O_ + o_red) * P_;

  const size_t chanStride = (size_t)P_ * G_;

  for (int p = wave; p < P_; p += 8) {          // uniform per wave: EXEC all-1
    // B fragments: 4x16 f32 B layout: lanes0-15 K={0,1}, lanes16-31 K={2,3}.
    v2f bfrag[4];
#pragma unroll
    for (int kc = 0; kc < 4; ++kc) {
      int k0 = 4 * kc + 2 * hf;
      const float* xp = x_pg + ((size_t)(b * C_ + k0) * P_ + p) * G_ + g0 + n;
      bfrag[kc].x = xp[0];
      bfrag[kc].y = xp[chanStride];
      // prefetch next iteration's rows (speculative; OOB at tail is dropped)
      __builtin_prefetch(xp + (size_t)8 * G_, 0, 3);
      __builtin_prefetch(xp + (size_t)8 * G_ + chanStride, 0, 3);
    }

    // y_p + bias + y_g combine: independent of WMMA, overlaps load latency
    float addv[2][8];
#pragma unroll
    for (int t = 0; t < 2; ++t)
#pragma unroll
      for (int r = 0; r < 8; ++r) {
        int o = 16 * t + 8 * hf + r;
        addv[t][r] = cadd[t][r] + y_p[(size_t)(b * O_ + o) * P_ + p];
      }

    v8f acc0 = {0.f, 0.f, 0.f, 0.f, 0.f, 0.f, 0.f, 0.f};
    v8f acc1 = {0.f, 0.f, 0.f, 0.f, 0.f, 0.f, 0.f, 0.f};
#pragma unroll
    for (int kc = 0; kc < 4; ++kc)
      acc0 = __builtin_amdgcn_wmma_f32_16x16x4_f32(false, afrag[0][kc],
                                                   false, bfrag[kc],
                                                   (short)0, acc0, false, false);
#pragma unroll
    for (int kc = 0; kc < 4; ++kc)
      acc1 = __builtin_amdgcn_wmma_f32_16x16x4_f32(false, afrag[1][kc],
                                                   false, bfrag[kc],
                                                   (short)0, acc1, false, false);

    // Epilogue: leaky-relu, store out_pg, running max over p, collect w[]
    float w[16];
#pragma unroll
    for (int t = 0; t < 2; ++t) {
#pragma unroll
      for (int r = 0; r < 8; ++r) {
        int o = 16 * t + 8 * hf + r;
        float acc = (t == 0) ? acc0[r] : acc1[r];
        float val = leaky(acc + addv[t][r]);                    // max(x,0.01x)
        out_pg[((size_t)(b * O_ + o) * P_ + p) * G_ + g0 + n] = val;
        ogmax[t][r] = fmaxf(ogmax[t][r], val);                  // max over p
        w[t * 8 + r] = val;
      }
    }

    // Register-combining butterfly: 16 regs x 16-lane max in 15 shuffles.
    // Round(mask m, half h): pair (w[i], w[i+h]); lanes with (lane&m)==0
    // carry the w[i] partial, lanes with (lane&m)!=0 carry w[i+h].
#pragma unroll
    for (int level = 0; level < 4; ++level) {
      int m = 1 << level;
      int h = 8 >> level;                       // 8,4,2,1
#pragma unroll
      for (int i = 0; i < h; ++i) {
        float a   = w[i];
        float bb  = w[i + h];
        float sel = (lane & m) ? bb : a;
        float oth = __shfl_xor(sel, m, 32);
        w[i] = fmaxf(sel, oth);
      }
    }
    // All 32 lanes: one atomic each, distinct o per lane, no divergence.
    atomicMax(opk_base + p, f2key(w[0]));
  }

  // out_g: this block exclusively owns (b, g0..g0+15); combine 8 waves in LDS.
  __shared__ unsigned ogk[O_ * 16];
  for (int i = threadIdx.x; i < O_ * 16; i += 256) ogk[i] = 0u;
  __syncthreads();
#pragma unroll
  for (int t = 0; t < 2; ++t)
#pragma unroll
    for (int r = 0; r < 8; ++r) {
      int o = 16 * t + 8 * hf + r;
      atomicMax(&ogk[o * 16 + n], f2key(ogmax[t][r]));
    }
  __syncthreads();
  for (int i = threadIdx.x; i < O_ * 16; i += 256) {
    int o = i >> 4, gi = i & 15;
    out_g[(size_t)(b * O_ + o) * G_ + g0 + gi] = key2f(ogk[i]);
  }
}

// ---------------- kernel 6: decode out_p keys ----------------
__global__ void decode_outp_k(const unsigned* __restrict__ keys,
                              float* __restrict__ out_p) {
  int i = blockIdx.x * 256 + threadIdx.x;           // < 32768
  out_p[i] = key2f(keys[i]);
}

extern "C" void kernel_launch(void* const* d_in, const int* in_sizes, int n_in,
                              void* d_out, int out_size, void* d_ws, size_t ws_size,
                              hipStream_t stream) {
  const float* x_pg    = (const float*)d_in[0];
  const float* x_p     = (const float*)d_in[1];
  const float* x_g     = (const float*)d_in[2];
  const float* W_pg_p  = (const float*)d_in[3];
  const float* b_pg_p  = (const float*)d_in[4];
  const float* W_pg_g  = (const float*)d_in[5];
  const float* b_pg_g  = (const float*)d_in[6];
  const float* W_pg_pg = (const float*)d_in[7];
  const float* b_pg_pg = (const float*)d_in[8];
  const float* W_p     = (const float*)d_in[9];
  const float* b_p     = (const float*)d_in[10];
  const float* W_g     = (const float*)d_in[11];
  const float* b_g     = (const float*)d_in[12];

  // workspace layout (4-byte units): total ~1.86 MB
  float*    ws        = (float*)d_ws;
  float*    pool_g    = ws;                         // 16384
  float*    pool_p    = ws + 16384;                 // 131072
  float*    y_p       = ws + 147456;                // 32768
  float*    y_g       = ws + 180224;                // 262144
  unsigned* outp_keys = (unsigned*)(ws + 442368);   // 32768

  float* out_pg = (float*)d_out;                            // B*O*P*G
  float* out_p  = out_pg + (size_t)B_ * O_ * P_ * G_;       // B*O*P
  float* out_g  = out_p + (size_t)B_ * O_ * P_;             // B*O*G

  init_keys_k<<<(B_ * O_ * P_) / 256, 256, 0, stream>>>(outp_keys);
  pool_g_k<<<(B_ * C_ * P_) / 8, 256, 0, stream>>>(x_pg, pool_g);
  pool_p_k<<<(B_ * C_ * G_) / 256, 256, 0, stream>>>(x_pg, pool_p);
  yp_k<<<(B_ * O_ * P_) / 256, 256, 0, stream>>>(pool_g, x_p, W_pg_p, b_pg_p,
                                                 W_p, b_p, y_p);
  yg_k<<<(B_ * O_ * G_) / 256, 256, 0, stream>>>(pool_p, x_g, W_pg_g, b_pg_g,
                                                 W_g, b_g, y_g);
  main_wmma_k<<<B_ * (G_ / 16), 256, 0, stream>>>(x_pg, W_pg_pg, b_pg_pg, y_p,
                                                  y_g, out_pg, outp_keys, out_g);
  decode_outp_k<<<(B_ * O_ * P_) / 256, 256, 0, stream>>>(outp_keys, out_p);
}